// Bilinear_15822659518756
// MI455X (gfx1250) — compile-verified
//
#include <hip/hip_runtime.h>
#include <hip/hip_bf16.h>
#include <stdint.h>

#define HDIM 224
#define WDIM 224
#define ROW_FLOATS (WDIM * 5)            // 1120 floats = 4480 bytes per (b,h) input row
#define ROW_BYTES  (ROW_FLOATS * 4)      // 4480
#define IN_CHUNKS  (ROW_BYTES / 16)      // 280 x b128 async load chunks
#define OROW_FLOATS (WDIM * 3)           // 672 floats = 2688 bytes per output row
#define OUT_CHUNKS  (OROW_FLOATS * 4 / 16) // 168 x b128 async store chunks

__global__ __launch_bounds__(WDIM) void warp_gather_kernel(
    const float* __restrict__ x, float* __restrict__ out)
{
    __shared__ float srow[ROW_FLOATS];   // staged input row (rgb+dx+dy per px)
    __shared__ float orow[OROW_FLOATS];  // packed output row (rgb per px)

    const int w = threadIdx.x;      // 0..223
    const int h = blockIdx.x;       // 0..223
    const int b = blockIdx.y;       // 0..B-1

    const size_t imgBase = (size_t)b * (HDIM * WDIM);     // pixels
    const size_t rowPix  = imgBase + (size_t)h * WDIM;    // pixel index of row start
    const float* rowBase = x + rowPix * 5;

    // ---- Stage this input row (4480B) into LDS with async b128 copies.
    // Thread t copies chunk t; threads 0..55 also copy chunk 224+t (280 total).
    {
        unsigned ldsOff = ((unsigned)(uintptr_t)&srow[0]) + (unsigned)(w * 16);
        const char* g0  = (const char*)rowBase + (size_t)w * 16u;
        asm volatile("global_load_async_to_lds_b128 %0, %1, off"
                     :: "v"(ldsOff), "v"(g0) : "memory");
        if (w < (IN_CHUNKS - WDIM)) {   // 56 tail chunks
            unsigned ldsOff1 = ldsOff + (unsigned)(WDIM * 16);
            const char* g1   = g0 + (size_t)(WDIM * 16);
            asm volatile("global_load_async_to_lds_b128 %0, %1, off"
                         :: "v"(ldsOff1), "v"(g1) : "memory");
        }
        asm volatile("s_wait_asynccnt 0x0" ::: "memory");
    }
    __syncthreads();

    // Warm L2 for the gather halo a few rows ahead of the stream (global_prefetch_b8).
    __builtin_prefetch(rowBase + 4 * ROW_FLOATS, 0, 3);

    // ---- dx,dy from LDS (stride-5 dwords across lanes -> conflict-free banks)
    const float dx = srow[w * 5 + 3];
    const float dy = srow[w * 5 + 4];

    // ---- floored mod into [0,224), then int + clamp (matches jnp.mod + JAX clamp)
    const float inv224 = 1.0f / 224.0f;
    float fx = (float)w + dx;
    float fy = (float)h + dy;
    fx = fmaf(-224.0f, floorf(fx * inv224), fx);
    fy = fmaf(-224.0f, floorf(fy * inv224), fy);
    int X = (int)fx;  X = X < 0 ? 0 : (X > WDIM - 1 ? WDIM - 1 : X);
    int Y = (int)fy;  Y = Y < 0 ? 0 : (Y > HDIM - 1 ? HDIM - 1 : Y);

    // ---- gather rgb through cache (source is within ~±5 px -> L0/L2 hits)
    const float* src = x + (imgBase + (size_t)Y * WDIM + (size_t)X) * 5;
    const float r  = src[0];
    const float g  = src[1];
    const float bl = src[2];

    // ---- pack output row in LDS (stride-3 dwords -> conflict-free banks)
    orow[w * 3 + 0] = r;
    orow[w * 3 + 1] = g;
    orow[w * 3 + 2] = bl;
    __syncthreads();

    // ---- stream the packed row out with coalesced async non-temporal b128 stores.
    // Row byte offset = rowPix*12 = 2688*k -> every 16B chunk is aligned.
    // ASYNCcnt drains via S_ENDPGM's implicit wait-idle; LDS lives until WG exit.
    if (w < OUT_CHUNKS) {
        unsigned ldsOff = ((unsigned)(uintptr_t)&orow[0]) + (unsigned)(w * 16);
        char* gout = (char*)out + rowPix * 12u + (size_t)w * 16u;
        asm volatile("global_store_async_from_lds_b128 %0, %1, off th:TH_STORE_NT"
                     :: "v"(gout), "v"(ldsOff) : "memory");
    }
}

extern "C" void kernel_launch(void* const* d_in, const int* in_sizes, int n_in,
                              void* d_out, int out_size, void* d_ws, size_t ws_size,
                              hipStream_t stream)
{
    (void)n_in; (void)out_size; (void)d_ws; (void)ws_size;
    const float* x = (const float*)d_in[0];
    float* out = (float*)d_out;

    const int B = in_sizes[0] / (HDIM * WDIM * 5);   // 256 for the reference shapes
    dim3 grid(HDIM, B);
    warp_gather_kernel<<<grid, WDIM, 0, stream>>>(x, out);
}